// Joint_29317446763071
// MI455X (gfx1250) — compile-verified
//
#include <hip/hip_runtime.h>
#include <stddef.h>

typedef float v2f __attribute__((ext_vector_type(2)));
typedef float v4f __attribute__((ext_vector_type(4)));
typedef float v8f __attribute__((ext_vector_type(8)));

#define D_DIM     512
#define V_DIM     1024
#define ENC_ROWS  2048   // 8*256
#define DEC_ROWS  512    // 8*64
#define TOT_ROWS  2560

// ---------------------------------------------------------------------------
// Stage 1: Y[r,v] = sum_d X[r,d] * W[v,d]   (X = concat(enc, dec), row-major)
//          bias[v] is added only to dec rows (r >= 2048) so the final
//          out = E + Dc contains bias exactly once.
// One wave computes a 16(M) x 64(N) tile via V_WMMA_F32_16X16X4_F32,
// 4 C-accumulators share each A fragment.
// ---------------------------------------------------------------------------
__device__ __forceinline__ void store_ctile(float* __restrict__ Y,
                                            const v8f& c, int mbase, int halo,
                                            int col, float badd)
{
#pragma unroll
    for (int i = 0; i < 8; ++i) {
        // C/D layout: VGPR i holds M = i (lanes 0-15) / M = i+8 (lanes 16-31)
        Y[(size_t)(mbase + i + halo * 8) * V_DIM + col] = c[i] + badd;
    }
}

__global__ void __launch_bounds__(128)
joint_gemm_wmma(const float* __restrict__ enc,
                const float* __restrict__ dec,
                const float* __restrict__ W,
                const float* __restrict__ bias,
                float* __restrict__ Y)
{
    const int lane = threadIdx.x & 31;
    const int wib  = threadIdx.x >> 5;
    const int w    = blockIdx.x * 4 + wib;   // global wave id, 0..2559
    const int mtile = w >> 4;                // 0..159  (16 rows each)
    const int ntile = w & 15;                // 0..15   (64 cols each)
    const int vbase = ntile * 64;

    const int halo = lane >> 4;              // 0: K even pair, 1: K odd pair
    const int l15  = lane & 15;

    // A fragment source: row = mtile*16 + (lane&15), K offset = 2*halo
    const int arow = mtile * 16 + l15;
    const float* aptr = (arow < ENC_ROWS)
                          ? (enc + (size_t)arow * D_DIM)
                          : (dec + (size_t)(arow - ENC_ROWS) * D_DIM);

    // B fragments: lane covers column v = vbase + j*16 + l15; W row-major [V,D]
    const float* bptr0 = W + (size_t)(vbase +  0 + l15) * D_DIM;
    const float* bptr1 = W + (size_t)(vbase + 16 + l15) * D_DIM;
    const float* bptr2 = W + (size_t)(vbase + 32 + l15) * D_DIM;
    const float* bptr3 = W + (size_t)(vbase + 48 + l15) * D_DIM;

    v8f c0 = {}, c1 = {}, c2 = {}, c3 = {};

#pragma unroll 4
    for (int k = 0; k < D_DIM; k += 4) {
        const int kk = k + 2 * halo;
        v2f a  = *(const v2f*)(aptr  + kk);
        v2f b0 = *(const v2f*)(bptr0 + kk);
        v2f b1 = *(const v2f*)(bptr1 + kk);
        v2f b2 = *(const v2f*)(bptr2 + kk);
        v2f b3 = *(const v2f*)(bptr3 + kk);
        // 8 args: (neg_a, A, neg_b, B, c_mod, C, reuse_a, reuse_b)
        c0 = __builtin_amdgcn_wmma_f32_16x16x4_f32(false, a, false, b0,
                                                   (short)0, c0, false, false);
        c1 = __builtin_amdgcn_wmma_f32_16x16x4_f32(false, a, false, b1,
                                                   (short)0, c1, false, false);
        c2 = __builtin_amdgcn_wmma_f32_16x16x4_f32(false, a, false, b2,
                                                   (short)0, c2, false, false);
        c3 = __builtin_amdgcn_wmma_f32_16x16x4_f32(false, a, false, b3,
                                                   (short)0, c3, false, false);
    }

    // Bias only into dec rows (tiles never straddle the 2048 boundary).
    const bool is_dec = (mtile >= ENC_ROWS / 16);
    const float bv0 = is_dec ? bias[vbase +  0 + l15] : 0.0f;
    const float bv1 = is_dec ? bias[vbase + 16 + l15] : 0.0f;
    const float bv2 = is_dec ? bias[vbase + 32 + l15] : 0.0f;
    const float bv3 = is_dec ? bias[vbase + 48 + l15] : 0.0f;

    const int mbase = mtile * 16;
    store_ctile(Y, c0, mbase, halo, vbase +  0 + l15, bv0);
    store_ctile(Y, c1, mbase, halo, vbase + 16 + l15, bv1);
    store_ctile(Y, c2, mbase, halo, vbase + 32 + l15, bv2);
    store_ctile(Y, c3, mbase, halo, vbase + 48 + l15, bv3);
}

// ---------------------------------------------------------------------------
// Stage 2: out[b,t,u,v] = E[b*256+t, v] + Dc[b*64+u, v]
// Pure streaming-store kernel: one float4 per thread, NT stores for the
// 512 MB output stream; E/Dc (10 MB) stay L2-resident.
// ---------------------------------------------------------------------------
__global__ void __launch_bounds__(256)
joint_bcast_add(const float* __restrict__ E,
                const float* __restrict__ Dc,
                float* __restrict__ out)
{
    const size_t tid  = (size_t)blockIdx.x * 256 + threadIdx.x;
    const int    v4   = (int)(tid & 255);        // 256 float4 per v-row
    const size_t rest = tid >> 8;
    const int    u    = (int)(rest & 63);
    const size_t bt   = rest >> 6;               // 0..2047
    const int    b    = (int)(bt >> 8);

    const v4f e = ((const v4f*)E )[bt * 256 + v4];
    const v4f d = ((const v4f*)Dc)[((size_t)(b * 64 + u)) * 256 + v4];
    v4f r = e + d;
    __builtin_nontemporal_store(r, ((v4f*)out) + tid);
}

// ---------------------------------------------------------------------------
extern "C" void kernel_launch(void* const* d_in, const int* in_sizes, int n_in,
                              void* d_out, int out_size, void* d_ws, size_t ws_size,
                              hipStream_t stream)
{
    const float* enc  = (const float*)d_in[0];   // (8,256,512)
    const float* dec  = (const float*)d_in[1];   // (8,64,512)
    const float* W    = (const float*)d_in[2];   // (1024,512)
    const float* bias = (const float*)d_in[3];   // (1024,)
    float*       out  = (float*)d_out;           // (8,256,64,1024)
    float*       Y    = (float*)d_ws;            // [2560,1024] fp32 = 10 MB

    // Stage 1: 160 mtiles * 16 ntiles = 2560 waves, 4 waves/block -> 640 blocks
    joint_gemm_wmma<<<640, 128, 0, stream>>>(enc, dec, W, bias, Y);

    const float* E  = Y;
    const float* Dc = Y + (size_t)ENC_ROWS * V_DIM;

    // Stage 2: 8*256*64*1024 floats / 4 = 33,554,432 float4 stores
    const size_t total4 = (size_t)8 * 256 * 64 * 256;
    joint_bcast_add<<<(unsigned)(total4 / 256), 256, 0, stream>>>(E, Dc, out);

    (void)in_sizes; (void)n_in; (void)out_size; (void)ws_size;
}